// EfficientMultiHop_53807350284457
// MI455X (gfx1250) — compile-verified
//
#include <hip/hip_runtime.h>
#include <stdint.h>

#define N_FEAT 64
#define FEAT2  32  // float2 elements per feature row
#define WAVES_PER_BLOCK 8

// ---- gfx1250 async-load-to-LDS path (guarded; shfl fallback otherwise) ----
#if defined(__AMDGCN__) && __has_builtin(__builtin_amdgcn_global_load_async_to_lds_b32)
#define USE_ASYNC_LDS 1
typedef __attribute__((address_space(1))) int gint_t;  // global AS int
typedef __attribute__((address_space(3))) int lint_t;  // LDS AS int
__device__ __forceinline__ gint_t* to_glb_i(const void* p) {
  return (gint_t*)(uintptr_t)p;
}
__device__ __forceinline__ lint_t* to_lds_i(void* p) {
  // generic LDS pointer: low 32 bits are the LDS byte offset
  return (lint_t*)(uint32_t)(uintptr_t)p;
}
__device__ __forceinline__ void wait_async0() {
#if __has_builtin(__builtin_amdgcn_s_wait_asynccnt)
  __builtin_amdgcn_s_wait_asynccnt(0);
#else
  asm volatile("s_wait_asynccnt 0x0" ::: "memory");
#endif
}
#else
#define USE_ASYNC_LDS 0
#endif

// ---------------- setup kernels ----------------

__global__ void gcn_zero_i32(int* __restrict__ p, int n) {
  int i = blockIdx.x * blockDim.x + threadIdx.x;
  if (i < n) p[i] = 0;
}

// histogram destinations (row) for CSR, sources (col) for degree
__global__ void gcn_count(const int* __restrict__ rowE,
                          const int* __restrict__ colE,
                          int* __restrict__ cntRow, int* __restrict__ cntCol, int E) {
  int e = blockIdx.x * blockDim.x + threadIdx.x;
  if (e >= E) return;
  atomicAdd(&cntRow[rowE[e]], 1);
  atomicAdd(&cntCol[colE[e]], 1);
}

// deg = (#col occurrences) + 1 (self loop); dis = rsqrt(deg), always > 0
__global__ void gcn_dis(const int* __restrict__ cntCol, float* __restrict__ dis, int N) {
  int i = blockIdx.x * blockDim.x + threadIdx.x;
  if (i < N) dis[i] = rsqrtf((float)(cntCol[i] + 1));
}

// single-block exclusive scan over N counts -> rowptr[N+1], cursor copy
__global__ void gcn_scan(const int* __restrict__ cnt, int* __restrict__ rowptr,
                         int* __restrict__ cursor, int n) {
  __shared__ int part[1024];
  int t = threadIdx.x;
  int chunk = (n + 1023) >> 10;
  int s0 = t * chunk;
  int s1 = s0 + chunk; if (s1 > n) s1 = n;
  int s = 0;
  for (int i = s0; i < s1; ++i) s += cnt[i];
  part[t] = s;
  __syncthreads();
  for (int off = 1; off < 1024; off <<= 1) {
    int v = (t >= off) ? part[t - off] : 0;
    __syncthreads();
    part[t] += v;
    __syncthreads();
  }
  int run = part[t] - s;  // exclusive base of this thread's chunk
  for (int i = s0; i < s1; ++i) { rowptr[i] = run; cursor[i] = run; run += cnt[i]; }
  if (t == 1023) rowptr[n] = part[1023];
}

// place each edge into its destination row's CSR segment; precompute norm
__global__ void gcn_scatter(const int* __restrict__ rowE,
                            const int* __restrict__ colE,
                            const float* __restrict__ dis,
                            int* __restrict__ cursor,
                            int* __restrict__ colIdx, float* __restrict__ normS, int E) {
  int e = blockIdx.x * blockDim.x + threadIdx.x;
  if (e >= E) return;
  int r = rowE[e];
  int c = colE[e];
  int pos = atomicAdd(&cursor[r], 1);
  colIdx[pos] = c;
  normS[pos]  = dis[r] * dis[c];
}

// ---------------- propagation ----------------
// One wave32 per destination node; lane owns float2 (feats 2*lane, 2*lane+1).
// Per 32-edge chunk the edge metadata is staged via global_load_async_to_lds
// (ASYNCcnt) and read back as LDS broadcasts; source feature rows are
// prefetched with global_prefetch_b8 before the dependent gather loop.
__device__ __forceinline__ float2 gcn_row_reduce(
    const float2* __restrict__ cur, const float* __restrict__ dis,
    const int* __restrict__ rowptr, const int* __restrict__ colIdx,
    const float* __restrict__ normS, int node, int lane,
    int* __restrict__ s_col, float* __restrict__ s_w) {
  float dr = dis[node];
  float2 v = cur[(size_t)node * FEAT2 + lane];
  float2 acc;
  acc.x = dr * dr * v.x;           // self-loop contribution (norm = dis^2)
  acc.y = dr * dr * v.y;
  int beg = rowptr[node];
  int end = rowptr[node + 1];
  for (int j0 = beg; j0 < end; j0 += 32) {
    int myj = j0 + lane;
    int cnt = end - j0; if (cnt > 32) cnt = 32;
#if USE_ASYNC_LDS
    if (myj < end) {
      __builtin_amdgcn_global_load_async_to_lds_b32(
          to_glb_i(colIdx + myj), to_lds_i(&s_col[lane]), 0, 0);
      __builtin_amdgcn_global_load_async_to_lds_b32(
          to_glb_i(normS + myj), to_lds_i(&s_w[lane]), 0, 0);
    }
    wait_async0();
    if (lane < cnt)
      __builtin_prefetch(cur + (size_t)s_col[lane] * FEAT2, 0, 3);  // global_prefetch_b8
    for (int k = 0; k < cnt; ++k) {
      int   c = s_col[k];     // uniform-address LDS broadcast
      float w = s_w[k];
      float2 s = cur[(size_t)c * FEAT2 + lane];   // coalesced 256B wave-load (L2 hot)
      acc.x += w * s.x;
      acc.y += w * s.y;
    }
#else
    int   cl = 0;
    float wl = 0.0f;
    if (myj < end) {
      cl = colIdx[myj];
      wl = normS[myj];
      __builtin_prefetch(cur + (size_t)cl * FEAT2, 0, 3);  // global_prefetch_b8
    }
    for (int k = 0; k < cnt; ++k) {
      int   c = __shfl(cl, k, 32);
      float w = __shfl(wl, k, 32);
      float2 s = cur[(size_t)c * FEAT2 + lane];
      acc.x += w * s.x;
      acc.y += w * s.y;
    }
#endif
  }
  return acc;
}

__global__ void gcn_hop1(const float* __restrict__ xf, const float* __restrict__ dis,
                         const int* __restrict__ rowptr, const int* __restrict__ colIdx,
                         const float* __restrict__ normS, float* __restrict__ y1f, int N) {
  __shared__ int   s_col[WAVES_PER_BLOCK][32];
  __shared__ float s_w  [WAVES_PER_BLOCK][32];
  int wslot = threadIdx.x >> 5;
  int node  = (blockIdx.x * blockDim.x + threadIdx.x) >> 5;
  int lane  = threadIdx.x & 31;
  if (node >= N) return;
  node = __builtin_amdgcn_readfirstlane(node);  // provably wave-uniform -> s_loads
  const float2* x = (const float2*)xf;
  float2 acc = gcn_row_reduce(x, dis, rowptr, colIdx, normS, node, lane,
                              s_col[wslot], s_w[wslot]);
  ((float2*)y1f)[(size_t)node * FEAT2 + lane] = acc;
}

// fused hop-2 + weighted combine: out = w0*x + w1*y1 + w2*(A_hat y1)
__global__ void gcn_final(const float* __restrict__ xf, const float* __restrict__ y1f,
                          const float* __restrict__ hw, const float* __restrict__ dis,
                          const int* __restrict__ rowptr, const int* __restrict__ colIdx,
                          const float* __restrict__ normS, float* __restrict__ outf, int N) {
  __shared__ int   s_col[WAVES_PER_BLOCK][32];
  __shared__ float s_w  [WAVES_PER_BLOCK][32];
  int wslot = threadIdx.x >> 5;
  int node  = (blockIdx.x * blockDim.x + threadIdx.x) >> 5;
  int lane  = threadIdx.x & 31;
  if (node >= N) return;
  node = __builtin_amdgcn_readfirstlane(node);
  // softmax over 3 hop weights (wave-uniform scalar work)
  float h0 = hw[0], h1 = hw[1], h2 = hw[2];
  float m  = fmaxf(h0, fmaxf(h1, h2));
  float e0 = __expf(h0 - m), e1 = __expf(h1 - m), e2 = __expf(h2 - m);
  float inv = 1.0f / (e0 + e1 + e2);
  float w0 = e0 * inv, w1 = e1 * inv, w2 = e2 * inv;

  const float2* y1 = (const float2*)y1f;
  float2 acc2 = gcn_row_reduce(y1, dis, rowptr, colIdx, normS, node, lane,
                               s_col[wslot], s_w[wslot]);
  float2 xv = ((const float2*)xf)[(size_t)node * FEAT2 + lane];
  float2 yv = y1[(size_t)node * FEAT2 + lane];
  float2 o;
  o.x = w0 * xv.x + w1 * yv.x + w2 * acc2.x;
  o.y = w0 * xv.y + w1 * yv.y + w2 * acc2.y;
  ((float2*)outf)[(size_t)node * FEAT2 + lane] = o;
}

// ---------------- launcher ----------------

extern "C" void kernel_launch(void* const* d_in, const int* in_sizes, int n_in,
                              void* d_out, int out_size, void* d_ws, size_t ws_size,
                              hipStream_t stream) {
  const float* x  = (const float*)d_in[0];
  const int*   ei = (const int*)d_in[1];     // int32 [2, E] (JAX x64 disabled)
  const float* hw = (const float*)d_in[2];   // 3 hop weights
  int N = in_sizes[0] / N_FEAT;
  int E = in_sizes[1] / 2;
  const int* rowE = ei;       // scatter destination
  const int* colE = ei + E;   // gather source

  // workspace carve-out (256B aligned)
  char* ws = (char*)d_ws;
  size_t off = 0;
  int*   cntRow = (int*)(ws + off);   off += (size_t)N * 4;        off = (off + 255) & ~(size_t)255;
  int*   cntCol = (int*)(ws + off);   off += (size_t)N * 4;        off = (off + 255) & ~(size_t)255;
  float* dis    = (float*)(ws + off); off += (size_t)N * 4;        off = (off + 255) & ~(size_t)255;
  int*   rowptr = (int*)(ws + off);   off += ((size_t)N + 1) * 4;  off = (off + 255) & ~(size_t)255;
  int*   cursor = (int*)(ws + off);   off += (size_t)N * 4;        off = (off + 255) & ~(size_t)255;
  int*   colIdx = (int*)(ws + off);   off += (size_t)E * 4;        off = (off + 255) & ~(size_t)255;
  float* normS  = (float*)(ws + off); off += (size_t)E * 4;        off = (off + 255) & ~(size_t)255;
  float* y1     = (float*)(ws + off); off += (size_t)N * N_FEAT * 4;
  (void)ws_size; (void)n_in; (void)out_size;

  const int tb = 256;
  int grN = (N + tb - 1) / tb;
  int grE = (E + tb - 1) / tb;
  int grW = (N * 32 + tb - 1) / tb;  // one wave32 per node

  gcn_zero_i32<<<grN, tb, 0, stream>>>(cntRow, N);
  gcn_zero_i32<<<grN, tb, 0, stream>>>(cntCol, N);
  gcn_count  <<<grE, tb, 0, stream>>>(rowE, colE, cntRow, cntCol, E);
  gcn_dis    <<<grN, tb, 0, stream>>>(cntCol, dis, N);
  gcn_scan   <<<1, 1024, 0, stream>>>(cntRow, rowptr, cursor, N);
  gcn_scatter<<<grE, tb, 0, stream>>>(rowE, colE, dis, cursor, colIdx, normS, E);
  gcn_hop1   <<<grW, tb, 0, stream>>>(x, dis, rowptr, colIdx, normS, y1, N);
  gcn_final  <<<grW, tb, 0, stream>>>(x, y1, hw, dis, rowptr, colIdx, normS, (float*)d_out, N);
}